// SupConLossWeighted_11785390260399
// MI455X (gfx1250) — compile-verified
//
#include <hip/hip_runtime.h>
#include <math.h>

#define NROWS 8192
#define DIM   128
#define TEMP_INV 10.0f
#define LOG_W (-0.69314718055994530942f)   // log(0.5)

typedef __attribute__((ext_vector_type(2))) float v2f;
typedef __attribute__((ext_vector_type(8))) float v8f;

// One wave (32 lanes) handles a 16-row anchor block.
// A-fragment (16x4 f32 per WMMA): lanes 0-15 hold M=lane with K=4k,4k+1 in .x/.y;
// lanes 16-31 hold the same rows with K=4k+2,4k+3.  B mirrors this with N=lane&15.
// C/D layout: VGPR v holds row (v + 8*hi), column = j0 + (lane&15).
__global__ __launch_bounds__(256)
void supcon_rows_kernel(const float* __restrict__ z,
                        const int*  __restrict__ labels,
                        float* __restrict__ partials) {
    const int lane   = threadIdx.x & 31;
    const int wave   = threadIdx.x >> 5;           // 0..7
    const int rowBlk = blockIdx.x * 8 + wave;      // 0..511
    const int i0     = rowBlk * 16;

    const int lg = lane & 15;                      // lane within half-wave
    const int hi = lane >> 4;                      // half-wave select

    // ---- resident A fragments: rows i0..i0+15, all K ----
    v2f a[32];
    {
        const float* arow = z + (size_t)(i0 + lg) * DIM + 2 * hi;
#pragma unroll
        for (int k = 0; k < 32; ++k) {
            a[k].x = arow[4 * k + 0];
            a[k].y = arow[4 * k + 1];
        }
    }

    int labRow[8];
#pragma unroll
    for (int v = 0; v < 8; ++v) labRow[v] = labels[i0 + v + 8 * hi];

    float runmax[8], runsum[8], consum[8];
    int   cnt[8];
#pragma unroll
    for (int v = 0; v < 8; ++v) {
        runmax[v] = -__builtin_inff();
        runsum[v] = 0.0f;
        consum[v] = 0.0f;
        cnt[v]    = 0;
    }

    for (int jt = 0; jt < NROWS / 16; ++jt) {
        const int   j0   = jt * 16;
        const float* brow = z + (size_t)(j0 + lg) * DIM + 2 * hi;

        v8f c = {};
#pragma unroll
        for (int g = 0; g < 4; ++g) {              // software-pipeline K in 4 groups
            v2f b[8];
#pragma unroll
            for (int k = 0; k < 8; ++k) {
                const int kk = g * 8 + k;
                b[k].x = brow[4 * kk + 0];
                b[k].y = brow[4 * kk + 1];
            }
#pragma unroll
            for (int k = 0; k < 8; ++k) {
                c = __builtin_amdgcn_wmma_f32_16x16x4_f32(
                        false, a[g * 8 + k], false, b[k],
                        (short)0, c, false, false);
            }
        }

        const int labC = labels[j0 + lg];
        const int gj   = j0 + lg;
#pragma unroll
        for (int v = 0; v < 8; ++v) {
            const int gi = i0 + v + 8 * hi;
            if (gi != gj) {                        // exclude diagonal (self pair)
                const bool  pos = (labRow[v] == labC);
                const float x   = c[v] * TEMP_INV;               // scaled similarity
                const float val = pos ? (x + LOG_W) : x;         // logsumexp input
                const float m   = fmaxf(runmax[v], val);
                runsum[v] = runsum[v] * __expf(runmax[v] - m) + __expf(val - m);
                runmax[v] = m;
                if (pos) { cnt[v]++; consum[v] += x; }
            }
        }
    }

    // ---- combine the 16 lanes of each half-wave (each holds 512 columns' partials) ----
#pragma unroll
    for (int v = 0; v < 8; ++v) {
#pragma unroll
        for (int m = 1; m <= 8; m <<= 1) {
            const float om  = __shfl_xor(runmax[v], m, 32);
            const float os  = __shfl_xor(runsum[v], m, 32);
            const int   oc  = __shfl_xor(cnt[v],    m, 32);
            const float ocs = __shfl_xor(consum[v], m, 32);
            const float M   = fmaxf(runmax[v], om);
            runsum[v] = runsum[v] * __expf(runmax[v] - M) + os * __expf(om - M);
            runmax[v] = M;
            cnt[v]   += oc;
            consum[v] += ocs;
        }
    }

    float part = 0.0f;
    if (lg == 0) {                                  // one lane per half-wave owns 8 rows
#pragma unroll
        for (int v = 0; v < 8; ++v) {
            if (cnt[v] > 0) {
                const float rep = runmax[v] + __logf(runsum[v]);
                part += rep - consum[v] / (float)cnt[v];   // rep + (-con) for this row
            }
        }
    }
    part += __shfl_xor(part, 16, 32);               // merge the two half-waves
    if (lane == 0) partials[rowBlk] = part;
}

// Fixed-tree deterministic reduction of the 512 per-wave partials.
__global__ __launch_bounds__(256)
void supcon_reduce_kernel(const float* __restrict__ partials,
                          float* __restrict__ out) {
    __shared__ float sm[256];
    const int t = threadIdx.x;
    sm[t] = partials[t] + partials[t + 256];
    __syncthreads();
#pragma unroll
    for (int off = 128; off > 0; off >>= 1) {
        if (t < off) sm[t] += sm[t + off];
        __syncthreads();
    }
    if (t == 0) out[0] = sm[0] / (float)NROWS;
}

extern "C" void kernel_launch(void* const* d_in, const int* in_sizes, int n_in,
                              void* d_out, int out_size, void* d_ws, size_t ws_size,
                              hipStream_t stream) {
    const float* z      = (const float*)d_in[0];
    const int*   labels = (const int*)d_in[1];
    float*       out    = (float*)d_out;
    float*       parts  = (float*)d_ws;            // 512 floats of scratch

    supcon_rows_kernel<<<64, 256, 0, stream>>>(z, labels, parts);
    supcon_reduce_kernel<<<1, 256, 0, stream>>>(parts, out);
}